// C_dht_26010321944863
// MI455X (gfx1250) — compile-verified
//
#include <hip/hip_runtime.h>
#include <math.h>
#include <stdint.h>
#include <string.h>

// Problem constants (fixed by the reference)
#define A_TOT  100            // angles
#define R_TOT  100            // rho bins
#define HDIM   100
#define WDIM   100
#define HW     10000
#define NC_TOT 2048           // N*C = 16*128
#define J_BLK  5              // angles per wave
#define N_AB   (A_TOT / J_BLK)        // 20 angle blocks
#define NC_SUB 8              // nc channels per wave
#define N_NCG  (NC_TOT / NC_SUB)      // 256 nc groups
#define WPB    4              // waves per block
#define HIST_F (NC_SUB * J_BLK * R_TOT)   // 4000 floats = 16000 B per wave
#define NCHUNK ((HW + 31) / 32)           // 313 pixel chunks of 32

#define TAB_FLOATS 200                     // cos'[100] ++ sin'[100]
#define TAB_BYTES  (TAB_FLOATS * 4)
#define FEATX_OFF  256                     // featx starts 1KB into d_ws (floats)

// ---- CDNA5 inline-asm helpers ------------------------------------------------

// LDS float atomic add, guaranteed ds_add_f32 (no CAS fallback), with
// compile-time immediate byte offset (per-nc slot inside the wave histogram).
template <int BYTE_OFF>
__device__ __forceinline__ void lds_atomic_add_f32(unsigned addr, float v) {
  asm volatile("ds_add_f32 %0, %1 offset:%2" :: "v"(addr), "v"(v), "i"(BYTE_OFF) : "memory");
}

// gfx1250 async-tensor path: LDS -> global, 16B per lane, tracked by ASYNCcnt.
__device__ __forceinline__ void async_store_b128(float* g, unsigned lds_byte) {
  asm volatile("global_store_async_from_lds_b128 %0, %1, off"
               :: "v"((unsigned long long)(uintptr_t)g), "v"(lds_byte) : "memory");
}

// ---- per-nc accumulate: one global load, 5 ds_add_f32 (rho addrs reused) -----
template <int NS>
__device__ __forceinline__ void nc_accum(const float* __restrict__ base, int off, bool ok,
                                         const unsigned (&a)[J_BLK]) {
  float v = base[NS * HW + off];       // saddr + 24-bit ioffset -> 1 instr
  v = ok ? v : 0.0f;                   // tail-chunk lanes add 0 to a valid bin
#pragma unroll
  for (int j = 0; j < J_BLK; ++j)
    lds_atomic_add_f32<NS * J_BLK * R_TOT * 4>(a[j], v);
}

// ---- main per-wave worker ----------------------------------------------------
// XM:    lanes traverse x (true) or y (false) -- pick the axis with the larger
//        rho slope so 32 consecutive lanes spread over ~16+ LDS banks.
// CONTIG: the selected traversal axis is contiguous in `src` (feat, or the
//        per-image transposed copy in workspace).
template <bool XM, bool CONTIG>
__device__ __forceinline__ void run_wave(const float* __restrict__ src, float* __restrict__ out,
                                         const float* __restrict__ tab,
                                         int nc0, int ab, int lane, float* histw, unsigned hb) {
  // Angle tables: host-precomputed float32(cos/sin(a*pi/100) in f64 / (142/99)),
  // bit-matching the numpy reference. 10 tiny loads instead of OCML f64 trig.
  float tc[J_BLK], ts[J_BLK];
  if (tab) {
#pragma unroll
    for (int j = 0; j < J_BLK; ++j) {
      tc[j] = tab[ab * J_BLK + j];
      ts[j] = tab[A_TOT + ab * J_BLK + j];
    }
  } else {
    // Emergency fallback only if d_ws cannot hold even the 800B table.
#pragma unroll
    for (int j = 0; j < J_BLK; ++j) {
      float ang = (float)(ab * J_BLK + j) * (float)(M_PI / (double)A_TOT);
      tc[j] = __cosf(ang) * (float)(99.0 / 142.0);
      ts[j] = __sinf(ang) * (float)(99.0 / 142.0);
    }
  }

  // Zero this wave's private histogram (no barrier needed: wave-private, and
  // DS ops from one wave stay in order).
  float4* h4 = (float4*)histw;
  for (int i = lane; i < HIST_F / 4; i += 32) h4[i] = make_float4(0.f, 0.f, 0.f, 0.f);

  const float* base = src + (size_t)nc0 * HW;
  for (int c = 0; c < NCHUNK; ++c) {
    int  q  = c * 32 + lane;          // linear index in traversal order
    bool ok = q < HW;
    int  qc = ok ? q : HW - 1;
    int  u  = qc / 100;               // slow axis
    int  vv = qc - u * 100;           // fast axis (lanes consecutive here)
    float cu = (float)(u  - 50);
    float cv = (float)(vv - 50);

    // rho bin byte-addresses for the 5 angles; exact numpy semantics:
    // separate f32 muls + f32 add (no FMA contraction) + round-half-to-even.
    unsigned a[J_BLK];
#pragma unroll
    for (int j = 0; j < J_BLK; ++j) {
      float xc = XM ? cv : cu;        // x is the fast axis in x-mode
      float yc = XM ? cu : cv;
      float rf = rintf(__fadd_rn(__fmul_rn(xc, tc[j]), __fmul_rn(yc, ts[j])));
      int   ri = (int)rf;             // in [-49, 49] by construction -> no clamp
      a[j] = (unsigned)((int)hb + j * (R_TOT * 4) + (R_TOT / 2) * 4 + (ri << 2));
    }

    int off = CONTIG ? qc : (vv * 100 + u);   // fallback: strided y-traversal
    nc_accum<0>(base, off, ok, a);
    nc_accum<1>(base, off, ok, a);
    nc_accum<2>(base, off, ok, a);
    nc_accum<3>(base, off, ok, a);
    nc_accum<4>(base, off, ok, a);
    nc_accum<5>(base, off, ok, a);
    nc_accum<6>(base, off, ok, a);
    nc_accum<7>(base, off, ok, a);
  }

  // Flush: make all ds_add_f32 results visible, then stream the histogram
  // straight LDS -> global with the gfx1250 async store path (no VGPR bounce).
  // hist[ns][j][r] (500 contiguous floats per ns) maps exactly onto
  // out[nc0+ns, ab*5 .. ab*5+4, 0..99] (also 500 contiguous floats).
  asm volatile("s_wait_dscnt 0" ::: "memory");
#pragma unroll
  for (int ns = 0; ns < NC_SUB; ++ns) {
    float*   g  = out + (size_t)(nc0 + ns) * (A_TOT * R_TOT) + (size_t)ab * (J_BLK * R_TOT);
    unsigned lb = hb + (unsigned)(ns * J_BLK * R_TOT * 4);
#pragma unroll
    for (int k = 0; k < 4; ++k) {
      int idx = k * 32 + lane;                    // 125 x 16B = 500 floats
      if (idx < (J_BLK * R_TOT) / 4) {
        async_store_b128(g + idx * 4, lb + (unsigned)idx * 16);
      }
    }
  }
#if __has_builtin(__builtin_amdgcn_s_wait_asynccnt)
  __builtin_amdgcn_s_wait_asynccnt(0);
#else
  asm volatile("s_wait_asynccnt 0" ::: "memory");
#endif
}

__global__ __launch_bounds__(WPB * 32) void dht_scatter(const float* __restrict__ feat,
                                                        const float* __restrict__ featx,
                                                        int featx_ok,
                                                        const float* __restrict__ tab,
                                                        float* __restrict__ out) {
  __shared__ __align__(16) float hist[WPB][HIST_F];   // 64000 B, 4 private 16KB slices
  const int lane = threadIdx.x & 31;
  const int w    = threadIdx.x >> 5;
  const int wid  = blockIdx.x * WPB + w;              // 0..5119
  const int ncg  = wid & (N_NCG - 1);                 // 0..255
  const int ab   = wid >> 8;                          // 0..19 angle block
  const int nc0  = ncg * NC_SUB;
  float*   histw = &hist[w][0];
  const unsigned hb = (unsigned)(uintptr_t)histw;     // LDS byte offset (low 32 of generic)

  // a in [0,25)u[75,100): |cos|>=|sin| -> traverse x; else traverse y.
  const bool xmode = (ab < 5) || (ab >= 15);
  if (xmode)          run_wave<true,  true >(feat,  out, tab, nc0, ab, lane, histw, hb);
  else if (featx_ok)  run_wave<false, true >(featx, out, tab, nc0, ab, lane, histw, hb);
  else                run_wave<false, false>(feat,  out, tab, nc0, ab, lane, histw, hb);
}

// One-time per-channel 100x100 image transpose into workspace so y-traversal
// waves get fully coalesced loads. ~164MB of L2-resident traffic, ~10us.
__global__ __launch_bounds__(256) void img_transpose(const float* __restrict__ in,
                                                     float* __restrict__ outT) {
  __shared__ float tile[32][33];
  const size_t base = (size_t)blockIdx.z * HW;
  const int x0 = blockIdx.x * 32, y0 = blockIdx.y * 32;
  for (int i = threadIdx.y; i < 32; i += 8) {
    int y = y0 + i, x = x0 + threadIdx.x;
    tile[i][threadIdx.x] = (y < HDIM && x < WDIM) ? in[base + y * WDIM + x] : 0.0f;
  }
  __syncthreads();
  for (int i = threadIdx.y; i < 32; i += 8) {
    int x = x0 + i, y = y0 + threadIdx.x;
    if (x < WDIM && y < HDIM) outT[base + x * HDIM + y] = tile[threadIdx.x][i];
  }
}

extern "C" void kernel_launch(void* const* d_in, const int* in_sizes, int n_in,
                              void* d_out, int out_size, void* d_ws, size_t ws_size,
                              hipStream_t stream) {
  (void)in_sizes; (void)n_in; (void)out_size;
  const float* feat = (const float*)d_in[0];
  float*       out  = (float*)d_out;
  float*       ws   = (float*)d_ws;

  // Host-side angle table, bit-matching the reference:
  //   tab_cos[a] = float32( cos(a * (pi/100)) / (142.0/99.0) )   (all in f64)
  // Static storage: the graph's H2D memcpy node re-reads it on every replay
  // with identical contents -> deterministic.
  static float h_tab[TAB_FLOATS];
  {
    const double itheta = M_PI / (double)A_TOT;
    const double irho   = 142.0 / 99.0;   // int(sqrt(100^2+100^2)+1) / (numrho-1)
    for (int a = 0; a < A_TOT; ++a) {
      h_tab[a]         = (float)(cos((double)a * itheta) / irho);
      h_tab[A_TOT + a] = (float)(sin((double)a * itheta) / irho);
    }
  }

  const int tab_ok = (ws_size >= (size_t)TAB_BYTES) ? 1 : 0;
  const int use_t  = (ws_size >= (size_t)FEATX_OFF * 4 + (size_t)NC_TOT * HW * 4) ? 1 : 0;
  const float* tab   = tab_ok ? ws : (const float*)nullptr;
  float*       featx = ws + FEATX_OFF;

  if (tab_ok) {
    hipMemcpyAsync((void*)ws, (const void*)h_tab, TAB_BYTES, hipMemcpyHostToDevice, stream);
  }
  if (use_t) {
    img_transpose<<<dim3((WDIM + 31) / 32, (HDIM + 31) / 32, NC_TOT),
                    dim3(32, 8), 0, stream>>>(feat, featx);
  }
  // 256 nc-groups x 20 angle-blocks = 5120 waves, 4 waves/block.
  // Every out[nc, a, r] is owned by exactly one wave -> no global atomics,
  // fully deterministic, d_out completely overwritten each call.
  dht_scatter<<<(N_NCG * N_AB) / WPB, WPB * 32, 0, stream>>>(feat, featx, use_t, tab, out);
}